// AdjacencyLayer_30605936951448
// MI455X (gfx1250) — compile-verified
//
#include <hip/hip_runtime.h>
#include <hip/hip_bf16.h>

typedef __attribute__((ext_vector_type(2))) float v2f;
typedef __attribute__((ext_vector_type(8))) float v8f;

#define Bn 1024
#define Dn 128

// Kernel 1: E = exp(1 - dc), norms[i] = sum_d x[i,d]^2
// 128 threads = 4 waves; one wave per row; 4 rows per block.
__global__ void adj_prep_kernel(const float* __restrict__ x,
                                const float* __restrict__ dc,
                                float* __restrict__ E,
                                float* __restrict__ norms) {
    const int wave = threadIdx.x >> 5;
    const int lane = threadIdx.x & 31;
    const int row  = blockIdx.x * 4 + wave;

    const float* xr = x  + row * Dn;
    const float* dr = dc + row * Dn;
    float*       er = E  + row * Dn;

    float s = 0.0f;
#pragma unroll
    for (int c = lane; c < Dn; c += 32) {
        float xv = xr[c];
        s += xv * xv;
        er[c] = expf(1.0f - dr[c]);
    }
    // wave32 butterfly reduction
#pragma unroll
    for (int off = 16; off > 0; off >>= 1)
        s += __shfl_xor(s, off, 32);
    if (lane == 0) norms[row] = s;
}

// Kernel 2: one wave per 16x16 output tile.
// S1 = X X^T, S2 = E E^T via V_WMMA_F32_16X16X4_F32 (exact f32 path),
// fused epilogue: sim = 1 / ((1-a)/D*(n_i + n_j - 2*S1) + (a/D)*S2), diag = 1.
__global__ void adj_wmma_kernel(const float* __restrict__ x,
                                const float* __restrict__ E,
                                const float* __restrict__ norms,
                                const float* __restrict__ dc_param,
                                float* __restrict__ out) {
    const int lane = threadIdx.x & 31;
    const int wave = threadIdx.x >> 5;
    const int tile = blockIdx.x * 8 + wave;   // 4096 tiles total
    const int trow = tile >> 6;               // tile / 64
    const int tcol = tile & 63;               // tile % 64
    const int row0 = trow * 16;
    const int col0 = tcol * 16;

    const int mn = lane & 15;   // M for A-frag, N for B-frag (per f32 layouts)
    const int hi = lane >> 4;   // lanes 16..31 carry K+2,K+3

    // 32-bit A 16x4 layout: lane m holds (k, k+1) [lo half] or (k+2, k+3) [hi half]
    const float* xa = x + (row0 + mn) * Dn + hi * 2;
    const float* xb = x + (col0 + mn) * Dn + hi * 2;  // B = X tile, row n, same K striping
    const float* ea = E + (row0 + mn) * Dn + hi * 2;
    const float* eb = E + (col0 + mn) * Dn + hi * 2;

    v8f acc1 = {};  // X X^T accumulator
    v8f acc2 = {};  // E E^T accumulator

#pragma unroll 4
    for (int k = 0; k < Dn; k += 4) {
        v2f av = *(const v2f*)(xa + k);
        v2f bv = *(const v2f*)(xb + k);
        acc1 = __builtin_amdgcn_wmma_f32_16x16x4_f32(
            /*neg_a=*/false, av, /*neg_b=*/false, bv,
            /*c_mod=*/(short)0, acc1, /*reuse_a=*/false, /*reuse_b=*/false);

        v2f aev = *(const v2f*)(ea + k);
        v2f bev = *(const v2f*)(eb + k);
        acc2 = __builtin_amdgcn_wmma_f32_16x16x4_f32(
            false, aev, false, bev, (short)0, acc2, false, false);
    }

    const float a  = dc_param[0];
    const float c1 = (1.0f - a) * (1.0f / (float)Dn);
    const float c2 = a * (1.0f / (float)Dn);

    const int   j  = col0 + mn;       // N index (lane)
    const float nj = norms[j];

    // f32 C/D layout: VGPR r holds M = r + 8*hi, N = lane&15
#pragma unroll
    for (int r = 0; r < 8; ++r) {
        const int   i     = row0 + r + hi * 8;
        const float denom = c1 * (norms[i] + nj - 2.0f * acc1[r]) + c2 * acc2[r];
        // v_rcp_f32 (~1 ulp) instead of the IEEE divide expansion; branchless
        // diagonal fix-up -> v_cndmask instead of an EXEC-masked branch.
        const float rv    = __builtin_amdgcn_rcpf(denom);
        out[i * Bn + j] = (i == j) ? 1.0f : rv;
    }
}

extern "C" void kernel_launch(void* const* d_in, const int* in_sizes, int n_in,
                              void* d_out, int out_size, void* d_ws, size_t ws_size,
                              hipStream_t stream) {
    const float* x        = (const float*)d_in[0];  // [1024,128] f32
    const float* dc       = (const float*)d_in[1];  // [1024,128] f32
    const float* dc_param = (const float*)d_in[2];  // [1] f32
    float*       out      = (float*)d_out;          // [1024,1024] f32

    float* E     = (float*)d_ws;      // 1024*128 f32 = 512 KB
    float* norms = E + Bn * Dn;       // 1024 f32

    adj_prep_kernel<<<Bn / 4, 128, 0, stream>>>(x, dc, E, norms);

    // 4096 tiles, 8 waves (tiles) per 256-thread block -> 512 blocks
    adj_wmma_kernel<<<(64 * 64) / 8, 256, 0, stream>>>(x, E, norms, dc_param, out);
}